// pointnet_sa_module_24532853195516
// MI455X (gfx1250) — compile-verified
//
#include <hip/hip_runtime.h>

// PointNet++ Set Abstraction for MI455X (gfx1250, wave32, WMMA).
// Precision: full fp32 end-to-end using V_WMMA_F32_16X16X4_F32 — the module is
// bandwidth/latency bound (~800MB activation traffic @23.3TB/s, sequential FPS),
// so fp32 matrix ops match reference numerics with no throughput cost.

#define B_      16
#define N_      4096
#define S_      1024
#define K_      32
#define CPTS    64
#define RAD2    0.04f
#define EPS_    1e-5f
#define M_TOTAL (B_ * S_ * K_)   // 524288 rows through the MLP

typedef float v2f __attribute__((ext_vector_type(2)));
typedef float v8f __attribute__((ext_vector_type(8)));

// ---------------------------------------------------------------------------
// 1) Farthest point sampling: one block per batch, xyz cached in LDS,
//    per-point running min-distance kept in registers (16 points/thread),
//    block argmax via wave shuffles + tiny LDS reduction.
// ---------------------------------------------------------------------------
__global__ void fps_kernel(const float* __restrict__ xyz,
                           float* __restrict__ newxyz) {
  extern __shared__ float sm[];
  float* px = sm;
  float* py = sm + N_;
  float* pz = sm + 2 * N_;
  float* red_d = sm + 3 * N_;            // 8 floats
  int*   red_i = (int*)(sm + 3 * N_ + 8); // 8 ints
  int*   sfar  = (int*)(sm + 3 * N_ + 16);

  const int b = blockIdx.x;
  const int tid = threadIdx.x;
  const float* bx = xyz + (size_t)b * N_ * 3;
  for (int i = tid; i < N_ * 3; i += 256) {
    float v = bx[i];
    int j = i / 3, c = i - j * 3;
    if (c == 0) px[j] = v; else if (c == 1) py[j] = v; else pz[j] = v;
  }
  __syncthreads();

  float dloc[16];
#pragma unroll
  for (int t = 0; t < 16; ++t) dloc[t] = 1e10f;

  int far = 0;
  for (int it = 0; it < S_; ++it) {
    const float fx = px[far], fy = py[far], fz = pz[far];
    if (tid == 0) {
      const size_t o = ((size_t)b * S_ + it) * 3;
      newxyz[o + 0] = fx; newxyz[o + 1] = fy; newxyz[o + 2] = fz;
    }
    float bd = -1.0f; int bi = 0;
#pragma unroll
    for (int t = 0; t < 16; ++t) {
      const int j = tid + (t << 8);
      const float dx = px[j] - fx, dy = py[j] - fy, dz = pz[j] - fz;
      const float d = dx * dx + dy * dy + dz * dz;
      const float nd = fminf(dloc[t], d);
      dloc[t] = nd;
      if (nd > bd || (nd == bd && j < bi)) { bd = nd; bi = j; }
    }
    // wave32 reduction (argmax, ties -> smallest index, matching jnp.argmax)
    for (int off = 16; off; off >>= 1) {
      const float od = __shfl_down(bd, off, 32);
      const int   oi = __shfl_down(bi, off, 32);
      if (od > bd || (od == bd && oi < bi)) { bd = od; bi = oi; }
    }
    if ((tid & 31) == 0) { red_d[tid >> 5] = bd; red_i[tid >> 5] = bi; }
    __syncthreads();
    if (tid == 0) {
      float cd = red_d[0]; int ci = red_i[0];
      for (int w = 1; w < 8; ++w)
        if (red_d[w] > cd || (red_d[w] == cd && red_i[w] < ci)) {
          cd = red_d[w]; ci = red_i[w];
        }
      *sfar = ci;
    }
    __syncthreads();
    far = *sfar;
  }
}

// ---------------------------------------------------------------------------
// 2) Ball query: one thread per centroid, batch xyz in LDS (broadcast reads),
//    sequential scan yields the first-K sorted indices exactly like the
//    reference's sort-and-slice. Wave early-exit once everyone has K.
// ---------------------------------------------------------------------------
__global__ void ballq_kernel(const float* __restrict__ xyz,
                             const float* __restrict__ newxyz,
                             int* __restrict__ bq) {
  extern __shared__ float sm[];
  float* px = sm; float* py = sm + N_; float* pz = sm + 2 * N_;
  const int b = blockIdx.x >> 2;
  const int s = ((blockIdx.x & 3) << 8) + threadIdx.x;
  const float* bx = xyz + (size_t)b * N_ * 3;
  for (int i = threadIdx.x; i < N_ * 3; i += 256) {
    float v = bx[i];
    int j = i / 3, c = i - j * 3;
    if (c == 0) px[j] = v; else if (c == 1) py[j] = v; else pz[j] = v;
  }
  __syncthreads();

  const size_t bs = (size_t)b * S_ + s;
  const float cx = newxyz[bs * 3 + 0];
  const float cy = newxyz[bs * 3 + 1];
  const float cz = newxyz[bs * 3 + 2];
  int cnt = 0, first = 0;
  int* dst = bq + bs * K_;
  for (int j = 0; j < N_; ++j) {
    const float dx = px[j] - cx, dy = py[j] - cy, dz = pz[j] - cz;
    const float d2 = dx * dx + dy * dy + dz * dz;
    if (d2 < RAD2) {
      if (cnt == 0) first = j;
      if (cnt < K_) dst[cnt] = j;
      ++cnt;
    }
    if (__ballot(cnt < K_) == 0ull) break;  // whole wave done
  }
  const int pad = (cnt > 0) ? first : 0;
  for (int k = (cnt < K_ ? cnt : K_); k < K_; ++k) dst[k] = pad;
}

__global__ void zero_stats_kernel(float* __restrict__ stats) {
  stats[threadIdx.x] = 0.0f;   // 256 floats: sum[0..127], sumsq[128..255]
}

// ---------------------------------------------------------------------------
// 3) GEMM + bias + BN-stats, fp32 WMMA 16x16x4.
//    Block = 256 threads = 8 waves; block tile = 128 rows x CO cols.
//    Whole K (padded to 68; stride 68 keeps LDS bank-conflict free) staged in
//    LDS once per block. GATHER variant builds layer-0 features
//    (centered xyz ++ grouped points) directly into the A tile.
// ---------------------------------------------------------------------------
template <int CO, bool GATHER>
__global__ void gemm_bn_kernel(const float* __restrict__ A,
                               const float* __restrict__ xyz,
                               const float* __restrict__ points,
                               const float* __restrict__ newxyz,
                               const int* __restrict__ bq,
                               const float* __restrict__ W,
                               const float* __restrict__ bias,
                               float* __restrict__ Z,
                               float* __restrict__ stats) {
  constexpr int CI  = GATHER ? 67 : 64;
  constexpr int LDK = 68;           // padded K stride (68 % 64 == 4 -> no conflicts)
  constexpr int NT  = CO / 16;
  extern __shared__ float sm[];
  float* lA   = sm;                 // 128 * 68
  float* lW   = lA + 128 * LDK;     // CO * 68 (zero padded)
  float* lB   = lW + CO * LDK;      // CO bias
  float* lsum = lB + CO;            // CO
  float* lsq  = lsum + CO;          // CO

  const int tid = threadIdx.x;
  const int m0  = blockIdx.x * 128;

  for (int i = tid; i < CO * LDK; i += 256) {
    const int o = i / LDK, c = i - o * LDK;
    lW[i] = (c < CI) ? W[o * CI + c] : 0.0f;
  }
  if (tid < CO) { lB[tid] = bias[tid]; lsum[tid] = 0.0f; lsq[tid] = 0.0f; }

  if constexpr (GATHER) {
    // row r = ((b*S + s)*K + k); two threads per row
    const int row  = tid >> 1;
    const int half = tid & 1;
    const int gr = m0 + row;
    const int b  = gr >> 15;              // S*K = 32768
    const int s  = (gr >> 5) & (S_ - 1);
    const int p  = bq[gr];
    const float* nx = newxyz + ((size_t)b * S_ + s) * 3;
    const float* qx = xyz    + ((size_t)b * N_ + p) * 3;
    const float* pp = points + ((size_t)b * N_ + p) * CPTS;
    float* dst = lA + row * LDK;
    if (half == 0) {
      dst[0] = qx[0] - nx[0];
      dst[1] = qx[1] - nx[1];
      dst[2] = qx[2] - nx[2];
#pragma unroll
      for (int c = 3; c < 34; ++c) dst[c] = pp[c - 3];
    } else {
#pragma unroll
      for (int c = 34; c < 67; ++c) dst[c] = pp[c - 3];
      dst[67] = 0.0f;
    }
  } else {
    for (int i = tid; i < 128 * 16; i += 256) {
      const int row = i >> 4, c4 = (i & 15) << 2;
      const float4 v = *reinterpret_cast<const float4*>(A + (size_t)(m0 + row) * 64 + c4);
      float* dst = lA + row * LDK + c4;
      dst[0] = v.x; dst[1] = v.y; dst[2] = v.z; dst[3] = v.w;
    }
    for (int i = tid; i < 128; i += 256) {
      float* dst = lA + i * LDK + 64;
      dst[0] = 0.0f; dst[1] = 0.0f; dst[2] = 0.0f; dst[3] = 0.0f;
    }
  }
  __syncthreads();

  const int lane = tid & 31;
  const int wv   = tid >> 5;     // wave id: 16-row strip
  const int lm   = lane & 15;
  const int hi   = lane >> 4;    // lane half selects K pair / M half

  v8f acc[NT];
  const v8f vz = {0, 0, 0, 0, 0, 0, 0, 0};
#pragma unroll
  for (int j = 0; j < NT; ++j) acc[j] = vz;

  // A frag (16x4, ISA layout): lane<16 holds M=lm K={k0,k0+1}; lane>=16 K={k0+2,k0+3}
  const float* aBase = lA + (wv * 16 + lm) * LDK + 2 * hi;
#pragma unroll
  for (int k0 = 0; k0 < LDK; k0 += 4) {
    const v2f a = *reinterpret_cast<const v2f*>(aBase + k0);
#pragma unroll
    for (int j = 0; j < NT; ++j) {
      // B frag (4x16): lane<16 N=lm K={k0,k0+1}; lane>=16 K={k0+2,k0+3}; B[k][n]=W[n][k]
      const v2f bb = *reinterpret_cast<const v2f*>(lW + (j * 16 + lm) * LDK + k0 + 2 * hi);
      acc[j] = __builtin_amdgcn_wmma_f32_16x16x4_f32(
          false, a, false, bb, (short)0, acc[j], false, false);
    }
  }

  // Epilogue: bias add, store z, per-channel sum/sumsq via LDS then global atomics.
  // C/D layout: VGPR i -> (M = wv*16 + 8*hi + i, N = j*16 + lm)
#pragma unroll
  for (int j = 0; j < NT; ++j) {
    const int n = j * 16 + lm;
    const float bn = lB[n];
    float ss = 0.0f, sq = 0.0f;
#pragma unroll
    for (int i = 0; i < 8; ++i) {
      const float v = acc[j][i] + bn;
      Z[(size_t)(m0 + wv * 16 + hi * 8 + i) * CO + n] = v;
      ss += v; sq += v * v;
    }
    atomicAdd(&lsum[n], ss);
    atomicAdd(&lsq[n], sq);
  }
  __syncthreads();
  if (tid < CO) {
    atomicAdd(&stats[tid], lsum[tid]);
    atomicAdd(&stats[128 + tid], lsq[tid]);
  }
}

// ---------------------------------------------------------------------------
// 4) BatchNorm (global mean/var over B*S*K, ddof=0) + ReLU, in place.
// ---------------------------------------------------------------------------
__global__ void bn_relu_kernel(float* __restrict__ Z, const float* __restrict__ stats,
                               const float* __restrict__ gamma,
                               const float* __restrict__ beta, int co) {
  const float invM = 1.0f / (float)M_TOTAL;
  const size_t total = (size_t)M_TOTAL * co;
  const int mask = co - 1;
  for (size_t i = (size_t)blockIdx.x * blockDim.x + threadIdx.x; i < total;
       i += (size_t)gridDim.x * blockDim.x) {
    const int c = (int)(i & mask);
    const float mean  = stats[c] * invM;
    const float var   = stats[128 + c] * invM - mean * mean;
    const float scale = gamma[c] * rsqrtf(var + EPS_);
    const float v = (Z[i] - mean) * scale + beta[c];
    Z[i] = v > 0.0f ? v : 0.0f;
  }
}

// ---------------------------------------------------------------------------
// 5) Max over K=32 neighbors -> new_points (B,S,128)
// ---------------------------------------------------------------------------
__global__ void maxpool_kernel(const float* __restrict__ Z, float* __restrict__ out) {
  const size_t total = (size_t)B_ * S_ * 128;
  for (size_t i = (size_t)blockIdx.x * blockDim.x + threadIdx.x; i < total;
       i += (size_t)gridDim.x * blockDim.x) {
    const size_t bs = i >> 7;
    const int c = (int)(i & 127);
    const float* src = Z + (bs * K_) * 128 + c;
    float m = src[0];
#pragma unroll
    for (int k = 1; k < K_; ++k) m = fmaxf(m, src[(size_t)k * 128]);
    out[i] = m;
  }
}

// ---------------------------------------------------------------------------
extern "C" void kernel_launch(void* const* d_in, const int* in_sizes, int n_in,
                              void* d_out, int out_size, void* d_ws, size_t ws_size,
                              hipStream_t stream) {
  (void)in_sizes; (void)n_in; (void)out_size; (void)ws_size;

  const float* xyz    = (const float*)d_in[0];
  const float* points = (const float*)d_in[1];
  const float* w0 = (const float*)d_in[2];
  const float* b0 = (const float*)d_in[3];
  const float* g0 = (const float*)d_in[4];
  const float* be0 = (const float*)d_in[5];
  const float* w1 = (const float*)d_in[6];
  const float* b1 = (const float*)d_in[7];
  const float* g1 = (const float*)d_in[8];
  const float* be1 = (const float*)d_in[9];
  const float* w2 = (const float*)d_in[10];
  const float* b2 = (const float*)d_in[11];
  const float* g2 = (const float*)d_in[12];
  const float* be2 = (const float*)d_in[13];

  float* out    = (float*)d_out;
  float* newxyz = out;                         // B*S*3
  float* newpts = out + (size_t)B_ * S_ * 3;   // B*S*128

  char* ws = (char*)d_ws;
  int*   bq    = (int*)ws;                                              // 2 MB
  float* stats = (float*)(ws + ((size_t)2 << 20));                      // 1 KB
  float* bufA  = (float*)(ws + ((size_t)4 << 20));                      // M*128 f32
  float* bufB  = (float*)(ws + ((size_t)4 << 20) + (size_t)M_TOTAL * 128 * 4); // M*64 f32

  const size_t smFps  = (size_t)(3 * N_ + 24) * sizeof(float);
  const size_t smBall = (size_t)(3 * N_) * sizeof(float);
  const size_t sm64   = (size_t)(128 * 68 + 64 * 68 + 3 * 64) * sizeof(float);
  const size_t sm128  = (size_t)(128 * 68 + 128 * 68 + 3 * 128) * sizeof(float);
  const int gblocks = M_TOTAL / 128;   // 4096

  fps_kernel<<<B_, 256, smFps, stream>>>(xyz, newxyz);
  ballq_kernel<<<B_ * (S_ / 256), 256, smBall, stream>>>(xyz, newxyz, bq);

  // Layer 0: fused gather (centered xyz ++ points) -> GEMM(67->64) + stats
  zero_stats_kernel<<<1, 256, 0, stream>>>(stats);
  gemm_bn_kernel<64, true><<<gblocks, 256, sm64, stream>>>(
      nullptr, xyz, points, newxyz, bq, w0, b0, bufA, stats);
  bn_relu_kernel<<<8192, 256, 0, stream>>>(bufA, stats, g0, be0, 64);

  // Layer 1: GEMM(64->64)
  zero_stats_kernel<<<1, 256, 0, stream>>>(stats);
  gemm_bn_kernel<64, false><<<gblocks, 256, sm64, stream>>>(
      bufA, nullptr, nullptr, nullptr, nullptr, w1, b1, bufB, stats);
  bn_relu_kernel<<<8192, 256, 0, stream>>>(bufB, stats, g1, be1, 64);

  // Layer 2: GEMM(64->128)
  zero_stats_kernel<<<1, 256, 0, stream>>>(stats);
  gemm_bn_kernel<128, false><<<gblocks, 256, sm128, stream>>>(
      bufB, nullptr, nullptr, nullptr, nullptr, w2, b2, bufA, stats);
  bn_relu_kernel<<<8192, 256, 0, stream>>>(bufA, stats, g2, be2, 128);

  maxpool_kernel<<<8192, 256, 0, stream>>>(bufA, newpts);
}